// L2VectorQuantizerKmeans_78408922956450
// MI455X (gfx1250) — compile-verified
//
#include <hip/hip_runtime.h>

// Shapes (compile-time constants from the reference)
#define BATCH    32
#define CH       256     // C_IN == CODE_DIM
#define HW       1024    // H*W
#define NCODE    1024
#define PLANE    (CH * HW)          // elements per batch image
#define OUT_ELEMS (BATCH * PLANE)   // 8388608

#define EPAD     260                // padded LDS row: 260 mod 64 == 4 -> bank-conflict-free b64 reads

typedef float v2f __attribute__((ext_vector_type(2)));
typedef float v8f __attribute__((ext_vector_type(8)));

// D = A(16x4) * B(4x16) + C, full fp32 (CDNA5 V_WMMA_F32_16X16X4_F32)
static __device__ __forceinline__ v8f wmma4(v2f a, v2f b, v8f c) {
    return __builtin_amdgcn_wmma_f32_16x16x4_f32(
        /*neg_a=*/false, a, /*neg_b=*/false, b,
        /*c_mod=*/(short)0, c, /*reuse_a=*/false, /*reuse_b=*/false);
}

// ---------------------------------------------------------------------------
// 1x1 conv as per-batch GEMM: Y[b] = W(256x256) @ X[b](256x1024) + bias
// One wave -> 16(M) x 64(N) strip.  8192 waves = 1024 blocks x 8 waves.
// Whole working set is L2-resident (192 MB) -> direct global operand streaming.
// ---------------------------------------------------------------------------
__global__ void conv1x1_wmma(const float* __restrict__ X,
                             const float* __restrict__ W,
                             const float* __restrict__ bias,
                             float* __restrict__ Y) {
    const int wave = (blockIdx.x * blockDim.x + threadIdx.x) >> 5;
    const int lane = threadIdx.x & 31;
    const int b    = wave >> 8;
    const int r    = wave & 255;
    const int m0   = (r >> 4) << 4;             // 0..240
    const int n0   = (r & 15) << 6;             // 0..960

    const float* __restrict__ Xb = X + (size_t)b * PLANE;
    float* __restrict__       Yb = Y + (size_t)b * PLANE;

    const int col  = lane & 15;
    const int hi2  = (lane >> 4) << 1;          // K offset 0 / 2 per half-wave
    const int mrow = m0 + col;

    v8f acc0 = {}, acc1 = {}, acc2 = {}, acc3 = {};
    #pragma unroll 4
    for (int k0 = 0; k0 < CH; k0 += 4) {
        v2f a;
        a.x = W[mrow * CH + k0 + hi2 + 0];
        a.y = W[mrow * CH + k0 + hi2 + 1];
        const float* Bp = Xb + (size_t)(k0 + hi2) * HW + n0 + col;
        v2f b0; b0.x = Bp[0];  b0.y = Bp[HW];
        v2f b1; b1.x = Bp[16]; b1.y = Bp[16 + HW];
        v2f b2; b2.x = Bp[32]; b2.y = Bp[32 + HW];
        v2f b3; b3.x = Bp[48]; b3.y = Bp[48 + HW];
        acc0 = wmma4(a, b0, acc0);
        acc1 = wmma4(a, b1, acc1);
        acc2 = wmma4(a, b2, acc2);
        acc3 = wmma4(a, b3, acc3);
    }

    const int rbase = m0 + ((lane >> 4) << 3);  // rows j / j+8 split
    #pragma unroll
    for (int j = 0; j < 8; ++j) {
        const int m = rbase + j;
        const float bv = bias[m];
        float* Yp = Yb + (size_t)m * HW + n0 + col;
        Yp[0]  = acc0[j] + bv;
        Yp[16] = acc1[j] + bv;
        Yp[32] = acc2[j] + bv;
        Yp[48] = acc3[j] + bv;
    }
}

// ---------------------------------------------------------------------------
// Codebook norms ||e_d||^2  (1024 codes x 256 dims)
// ---------------------------------------------------------------------------
__global__ void enorm_kernel(const float* __restrict__ E, float* __restrict__ en) {
    const int d = blockIdx.x * blockDim.x + threadIdx.x;
    if (d < NCODE) {
        float s = 0.f;
        #pragma unroll 8
        for (int c = 0; c < CH; ++c) { const float v = E[d * CH + c]; s += v * v; }
        en[d] = s;
    }
}

// ---------------------------------------------------------------------------
// Fused distance GEMM + argmin + gather + loss.
//   * Each wave owns 16 pixels; their 256-ch vectors live in 128 VGPRs for
//     the whole kernel (loaded once, reused for all 64 code tiles).
//   * Embedding tiles (16 codes x 256 ch) double-buffered in LDS, shared by
//     the 8 waves of the block; rows padded to 260 floats -> conflict-free
//     ds_load_b64 A-fragments.
//   * Inner loop: 1 ds_load_b64 + 1 v_wmma_f32_16x16x4_f32 per k-step.
// dist = ||e||^2 - 2 * (e . z)   (||z||^2 constant per pixel -> dropped)
// ---------------------------------------------------------------------------
__global__ void __launch_bounds__(256)
vq_kernel(const float* __restrict__ zc,    // z_conv [B,256,1024]
          const float* __restrict__ E,     // [1024,256]
          const float* __restrict__ en,    // [1024]
          float* __restrict__ zq,          // [B,256,1024]
          float* __restrict__ loss) {
    __shared__ float Et[2][16 * EPAD];

    const int lane = threadIdx.x & 31;
    const int wid  = threadIdx.x >> 5;                  // 0..7
    const int b    = blockIdx.x >> 3;                   // 8 blocks per batch
    const int n0   = ((blockIdx.x & 7) << 7) + (wid << 4);

    const float* __restrict__ zcb = zc + (size_t)b * PLANE;
    const int col = lane & 15;
    const int hi2 = (lane >> 4) << 1;
    const int hi8 = (lane >> 4) << 3;

    // ---- load this wave's 16-pixel B-operand into registers (64 v2f) ----
    v2f Bfrag[64];
    #pragma unroll
    for (int i = 0; i < 64; ++i) {
        const float* p = zcb + (size_t)(4 * i + hi2) * HW + n0 + col;
        Bfrag[i].x = p[0];
        Bfrag[i].y = p[HW];
    }

    // ---- stage embedding tile 0 ----
    {
        const float2* Eg = (const float2*)(E);
        for (int i = threadIdx.x; i < 2048; i += 256) {
            const int row = i >> 7;                     // 128 float2 per code row
            const int kk  = (i & 127) << 1;
            const float2 v = Eg[row * 128 + (i & 127)];
            Et[0][row * EPAD + kk]     = v.x;
            Et[0][row * EPAD + kk + 1] = v.y;
        }
    }
    __syncthreads();

    float minv = 3.402823466e38f;
    int   mini = 0;

    for (int t = 0; t < 64; ++t) {
        const int cur = t & 1;
        // prefetch next embedding tile into the other buffer
        if (t + 1 < 64) {
            const float2* Eg = (const float2*)(E + (size_t)(t + 1) * 16 * CH);
            float* dst = Et[cur ^ 1];
            for (int i = threadIdx.x; i < 2048; i += 256) {
                const int row = i >> 7;
                const int kk  = (i & 127) << 1;
                const float2 v = Eg[row * 128 + (i & 127)];
                dst[row * EPAD + kk]     = v.x;
                dst[row * EPAD + kk + 1] = v.y;
            }
        }

        // ---- 16x16 distance tile: codes [16t,16t+16) x this wave's pixels ----
        v8f acc = {};
        const float* Ebuf = Et[cur] + col * EPAD + hi2;
        #pragma unroll
        for (int i = 0; i < 64; ++i) {
            const v2f a = *(const v2f*)(Ebuf + 4 * i);
            acc = wmma4(a, Bfrag[i], acc);
        }

        const int mbase = (t << 4) + hi8;
        #pragma unroll
        for (int j = 0; j < 8; ++j) {
            const int m = mbase + j;
            const float d = en[m] - 2.0f * acc[j];
            if (d < minv) { minv = d; mini = m; }       // m monotonic -> first-min rule
        }
        __syncthreads();                                // buffer cur^1 staged; cur free next iter
    }

    // merge half-waves (rows j in lanes<16, rows j+8 in lanes>=16)
    {
        const float ov = __shfl_xor(minv, 16, 32);
        const int   oi = __shfl_xor(mini, 16, 32);
        if (ov < minv || (ov == minv && oi < mini)) { minv = ov; mini = oi; }
    }
    // winning code index of this lane's pixel column
    const int myidx = __shfl(mini, col, 32);

    // gather embedding[idx] -> zq; SSE from register-resident z_conv fragments
    float sse = 0.f;
    float* __restrict__ zqb = zq + (size_t)b * PLANE;
    #pragma unroll 8
    for (int i = 0; i < 64; ++i) {
        const int c = 4 * i + hi2;
        const v2f ev = *(const v2f*)(E + (size_t)myidx * CH + c);
        const float d0 = ev.x - Bfrag[i].x;
        const float d1 = ev.y - Bfrag[i].y;
        sse += d0 * d0 + d1 * d1;
        float* q = zqb + (size_t)c * HW + n0 + col;
        q[0]  = ev.x;
        q[HW] = ev.y;
    }
    // wave32 reduction, then one fp32 global atomic
    #pragma unroll
    for (int off = 16; off >= 1; off >>= 1) sse += __shfl_xor(sse, off, 32);
    if (lane == 0)
        atomicAdd(loss, sse * ((1.0f + 0.25f) / (float)OUT_ELEMS));
}

// ---------------------------------------------------------------------------
extern "C" void kernel_launch(void* const* d_in, const int* in_sizes, int n_in,
                              void* d_out, int out_size, void* d_ws, size_t ws_size,
                              hipStream_t stream) {
    (void)in_sizes; (void)n_in; (void)out_size; (void)ws_size;
    const float* z  = (const float*)d_in[0];
    const float* qw = (const float*)d_in[1];
    const float* qb = (const float*)d_in[2];
    const float* pw = (const float*)d_in[3];
    const float* pb = (const float*)d_in[4];
    const float* E  = (const float*)d_in[5];

    float* out   = (float*)d_out;            // [B,256,H,W]
    float* loss  = out + OUT_ELEMS;          // scalar
    float* zconv = loss + 1;                 // [B,256,H,W]
    float* zq    = zconv + OUT_ELEMS;        // [B,256,H,W]
    float* enorm = (float*)d_ws;             // 1024 floats scratch

    hipMemsetAsync(loss, 0, sizeof(float), stream);
    enorm_kernel<<<NCODE / 256, 256, 0, stream>>>(E, enorm);
    conv1x1_wmma<<<1024, 256, 0, stream>>>(z, qw, qb, zconv);
    vq_kernel<<<256, 256, 0, stream>>>(zconv, E, enorm, zq, loss);
    conv1x1_wmma<<<1024, 256, 0, stream>>>(zq, pw, pb, out);
}